// DyGrEncoder_Model_8581344657593
// MI455X (gfx1250) — compile-verified
//
#include <hip/hip_runtime.h>

#define NN 50000
#define NE 1600000
#define FD 32
#define NL 32
#define MT (NN / 16)            // 3125 row tiles, exact
#define SCAN_B 256
#define NBLK ((NN + SCAN_B - 1) / SCAN_B)   // 196 scan blocks

typedef __attribute__((ext_vector_type(2))) float v2f;
typedef __attribute__((ext_vector_type(8))) float v8f;

__device__ __forceinline__ v8f wmma4(v2f a, v2f b, v8f c) {
  // D = A(16x4 f32) * B(4x16 f32) + C(16x16 f32)
  return __builtin_amdgcn_wmma_f32_16x16x4_f32(false, a, false, b, (short)0, c,
                                               false, false);
}

__device__ __forceinline__ float sigm(float x) { return 1.0f / (1.0f + __expf(-x)); }

// ---------------- CSR build: degree count ----------------
__global__ void zero_i_kernel(int* __restrict__ p, int n) {
  int i = blockIdx.x * blockDim.x + threadIdx.x;
  if (i < n) p[i] = 0;
}

__global__ void deg_kernel(const int* __restrict__ dst, int* __restrict__ deg) {
  int e = blockIdx.x * blockDim.x + threadIdx.x;
  if (e < NE) atomicAdd(deg + dst[e], 1);
}

// ---------------- CSR build: 3-phase exclusive scan (LDS Hillis-Steele) ----------------
__global__ void scan_local_kernel(const int* __restrict__ deg,
                                  int* __restrict__ rowptr,
                                  int* __restrict__ bsum) {
  __shared__ int tmp[SCAN_B];
  int i = blockIdx.x * SCAN_B + threadIdx.x;
  int v = (i < NN) ? deg[i] : 0;
  tmp[threadIdx.x] = v;
  __syncthreads();
  for (int off = 1; off < SCAN_B; off <<= 1) {
    int t = (threadIdx.x >= off) ? tmp[threadIdx.x - off] : 0;
    __syncthreads();
    tmp[threadIdx.x] += t;
    __syncthreads();
  }
  if (i < NN) rowptr[i] = tmp[threadIdx.x] - v;   // exclusive within block
  if (threadIdx.x == SCAN_B - 1) bsum[blockIdx.x] = tmp[SCAN_B - 1];
}

__global__ void scan_bsum_kernel(int* __restrict__ bsum) {  // single block, NBLK <= 256
  __shared__ int tmp[SCAN_B];
  int v = (threadIdx.x < NBLK) ? bsum[threadIdx.x] : 0;
  tmp[threadIdx.x] = v;
  __syncthreads();
  for (int off = 1; off < SCAN_B; off <<= 1) {
    int t = (threadIdx.x >= off) ? tmp[threadIdx.x - off] : 0;
    __syncthreads();
    tmp[threadIdx.x] += t;
    __syncthreads();
  }
  if (threadIdx.x < NBLK) bsum[threadIdx.x] = tmp[threadIdx.x] - v;  // exclusive
}

__global__ void scan_add_kernel(int* __restrict__ rowptr, const int* __restrict__ bsum) {
  int i = blockIdx.x * SCAN_B + threadIdx.x;
  if (i < NN) rowptr[i] += bsum[blockIdx.x];
  if (i == 0) rowptr[NN] = NE;
}

// ---------------- CSR build: scatter edges, fold mean (ew / max(deg,1)) into weight ----
__global__ void scatter_kernel(const int* __restrict__ ei,    // (2,E): src then dst
                               const float* __restrict__ ew,
                               const int* __restrict__ deg,
                               const int* __restrict__ rowptr,
                               int* __restrict__ cursor,
                               int* __restrict__ csr_src,
                               float* __restrict__ csr_w) {
  int e = blockIdx.x * blockDim.x + threadIdx.x;
  if (e >= NE) return;
  int s = ei[e];
  int d = ei[NE + e];
  int pos = rowptr[d] + atomicAdd(cursor + d, 1);
  csr_src[pos] = s;
  csr_w[pos] = ew[e] / fmaxf((float)deg[d], 1.0f);
}

// ---------------- m = h @ W (NNx32 @ 32x32), fp32 WMMA ----------------
__global__ void gemm_m_kernel(const float* __restrict__ h,
                              const float* __restrict__ W,   // (32,32) row-major [k][j]
                              float* __restrict__ m) {
  int wave = (blockIdx.x * blockDim.x + threadIdx.x) >> 5;
  if (wave >= MT) return;
  int lane = threadIdx.x & 31;
  int lo = lane & 15;
  int hi = lane >> 4;
  int arow = wave * 16 + lo;
  v8f c0 = {}; v8f c1 = {};
#pragma unroll
  for (int k0 = 0; k0 < FD; k0 += 4) {
    int k = k0 + 2 * hi;
    v2f a = *(const v2f*)(h + (size_t)arow * FD + k);
    v2f b0, b1;
    b0.x = W[(k + 0) * FD + lo];      b0.y = W[(k + 1) * FD + lo];
    b1.x = W[(k + 0) * FD + 16 + lo]; b1.y = W[(k + 1) * FD + 16 + lo];
    c0 = wmma4(a, b0, c0);
    c1 = wmma4(a, b1, c1);
  }
#pragma unroll
  for (int v = 0; v < 8; ++v) {
    int r = wave * 16 + v + 8 * hi;
    m[(size_t)r * FD + lo]      = c0[v];
    m[(size_t)r * FD + 16 + lo] = c1[v];
  }
}

// ---------------- gather-reduce: agg[n] = sum_{e in CSR row n} w_e * m[src_e]  --------
// one wave per node; lane = feature; indices loaded coalesced then shfl-broadcast
__global__ void gather_kernel(const int* __restrict__ rowptr,
                              const int* __restrict__ csr_src,
                              const float* __restrict__ csr_w,
                              const float* __restrict__ m,
                              float* __restrict__ agg) {
  int node = (blockIdx.x * blockDim.x + threadIdx.x) >> 5;
  if (node >= NN) return;
  int lane = threadIdx.x & 31;
  int beg = rowptr[node];
  int end = rowptr[node + 1];
  float acc = 0.0f;
  for (int base = beg; base < end; base += 32) {
    int idx = base + lane;
    int s = 0;
    float w = 0.0f;
    if (idx < end) { s = csr_src[idx]; w = csr_w[idx]; }
    int cnt = min(32, end - base);
    for (int j = 0; j < cnt; ++j) {
      int   sj = __shfl(s, j, 32);
      float wj = __shfl(w, j, 32);
      acc += wj * m[(size_t)sj * FD + lane];
    }
  }
  agg[(size_t)node * FD + lane] = acc;
}

// ---------------- fused GRU cell (agg already mean-scaled) ----------------
__global__ void gru_kernel(const float* __restrict__ h,
                           const float* __restrict__ agg,
                           const float* __restrict__ wih,   // (96,32)
                           const float* __restrict__ whh,   // (96,32)
                           const float* __restrict__ bih,   // (96)
                           const float* __restrict__ bhh,   // (96)
                           float* __restrict__ hout) {
  int wave = (blockIdx.x * blockDim.x + threadIdx.x) >> 5;
  if (wave >= MT) return;
  int lane = threadIdx.x & 31;
  int lo = lane & 15;
  int hi = lane >> 4;
  int arow = wave * 16 + lo;

  v8f gi[6], gh[6];
#pragma unroll
  for (int t = 0; t < 6; ++t) {
    float bi = bih[t * 16 + lo];
    float bh = bhh[t * 16 + lo];
#pragma unroll
    for (int v = 0; v < 8; ++v) { gi[t][v] = bi; gh[t][v] = bh; }
  }

#pragma unroll
  for (int k0 = 0; k0 < FD; k0 += 4) {
    int k = k0 + 2 * hi;
    v2f aagg = *(const v2f*)(agg + (size_t)arow * FD + k);
    v2f ah   = *(const v2f*)(h + (size_t)arow * FD + k);
#pragma unroll
    for (int t = 0; t < 6; ++t) {
      int col = t * 16 + lo;
      v2f bi = *(const v2f*)(wih + (size_t)col * FD + k);  // B[k][col] = wih[col][k]
      v2f bh = *(const v2f*)(whh + (size_t)col * FD + k);
      gi[t] = wmma4(aagg, bi, gi[t]);
      gh[t] = wmma4(ah,   bh, gh[t]);
    }
  }

  // gates: r -> tiles 0,1 ; z -> tiles 2,3 ; n -> tiles 4,5
#pragma unroll
  for (int p = 0; p < 2; ++p) {
#pragma unroll
    for (int v = 0; v < 8; ++v) {
      int node = wave * 16 + v + 8 * hi;
      int j = p * 16 + lo;
      float hold = h[(size_t)node * FD + j];
      float r = sigm(gi[p][v] + gh[p][v]);
      float z = sigm(gi[2 + p][v] + gh[2 + p][v]);
      float n = tanhf(gi[4 + p][v] + r * gh[4 + p][v]);
      hout[(size_t)node * FD + j] = (1.0f - z) * n + z * hold;
    }
  }
}

// ---------------- LSTM (seq_len=1, h0=c0=0): out = sig(o)*tanh(sig(i)*tanh(g)) --------
__global__ void lstm_kernel(const float* __restrict__ h,
                            const float* __restrict__ wih,  // (128,32)
                            const float* __restrict__ bih,  // (128)
                            const float* __restrict__ bhh,  // (128)
                            float* __restrict__ hout) {
  int wave = (blockIdx.x * blockDim.x + threadIdx.x) >> 5;
  if (wave >= MT) return;
  int lane = threadIdx.x & 31;
  int lo = lane & 15;
  int hi = lane >> 4;
  int arow = wave * 16 + lo;

  const int tiles[6] = {0, 1, 4, 5, 6, 7};  // i, g, o column tiles (f unused: c0 == 0)
  v8f acc[6];
#pragma unroll
  for (int t = 0; t < 6; ++t) {
    float b = bih[tiles[t] * 16 + lo] + bhh[tiles[t] * 16 + lo];
#pragma unroll
    for (int v = 0; v < 8; ++v) acc[t][v] = b;
  }

#pragma unroll
  for (int k0 = 0; k0 < FD; k0 += 4) {
    int k = k0 + 2 * hi;
    v2f a = *(const v2f*)(h + (size_t)arow * FD + k);
#pragma unroll
    for (int t = 0; t < 6; ++t) {
      int col = tiles[t] * 16 + lo;
      v2f b = *(const v2f*)(wih + (size_t)col * FD + k);
      acc[t] = wmma4(a, b, acc[t]);
    }
  }

#pragma unroll
  for (int p = 0; p < 2; ++p) {
#pragma unroll
    for (int v = 0; v < 8; ++v) {
      int node = wave * 16 + v + 8 * hi;
      float ii = sigm(acc[p][v]);
      float gg = tanhf(acc[2 + p][v]);
      float oo = sigm(acc[4 + p][v]);
      float c = ii * gg;
      hout[(size_t)node * FD + p * 16 + lo] = oo * tanhf(c);
    }
  }
}

// ---------------- final: out[n] = relu(h[n,:]) . lin_w + lin_b ----------------
__global__ void final_kernel(const float* __restrict__ h,
                             const float* __restrict__ lin_w,
                             const float* __restrict__ lin_b,
                             float* __restrict__ out) {
  int node = (blockIdx.x * blockDim.x + threadIdx.x) >> 5;
  if (node >= NN) return;
  int lane = threadIdx.x & 31;
  float v = fmaxf(h[(size_t)node * FD + lane], 0.0f) * lin_w[lane];
#pragma unroll
  for (int off = 16; off > 0; off >>= 1) v += __shfl_xor(v, off, 32);
  if (lane == 0) out[node] = v + lin_b[0];
}

extern "C" void kernel_launch(void* const* d_in, const int* in_sizes, int n_in,
                              void* d_out, int out_size, void* d_ws, size_t ws_size,
                              hipStream_t stream) {
  (void)in_sizes; (void)n_in; (void)out_size; (void)ws_size;
  const float* x  = (const float*)d_in[0];
  const int*   ei = (const int*)d_in[1];
  const float* ew = (const float*)d_in[2];
  const float* p[23];
  for (int i = 0; i < 23; ++i) p[i] = (const float*)d_in[i];

  const size_t NF = (size_t)NN * FD;
  float* ws    = (float*)d_ws;
  float* hA    = ws + 0 * NF;
  float* hB    = ws + 1 * NF;
  float* mb    = ws + 2 * NF;          // per-step m; also block-2 LSTM output
  float* agg   = ws + 3 * NF;
  float* hC    = ws + 4 * NF;          // block-1 LSTM output
  float* csr_w = ws + 5 * NF;          // NE floats
  int*   iw      = (int*)(ws + 5 * NF + NE);
  int*   deg     = iw;                 // NN
  int*   rowptr  = iw + NN;            // NN + 1
  int*   cursor  = iw + 2 * NN + 1;    // NN
  int*   bsum    = iw + 3 * NN + 1;    // 256
  int*   csr_src = iw + 3 * NN + 1 + 256;  // NE

  dim3 B(256);
  dim3 gN((NN + 255) / 256);           // thread per node
  dim3 gE((NE + 255) / 256);           // thread per edge
  dim3 gT((MT * 32 + 255) / 256);      // wave per 16-node tile
  dim3 gNw((NN * 32 + 255) / 256);     // wave per node

  // ---- build mean-weighted CSR once (reused by all 64 propagation steps) ----
  zero_i_kernel<<<gN, B, 0, stream>>>(deg, NN);
  zero_i_kernel<<<gN, B, 0, stream>>>(cursor, NN);
  deg_kernel<<<gE, B, 0, stream>>>(ei + NE, deg);
  scan_local_kernel<<<NBLK, SCAN_B, 0, stream>>>(deg, rowptr, bsum);
  scan_bsum_kernel<<<1, SCAN_B, 0, stream>>>(bsum);
  scan_add_kernel<<<NBLK, SCAN_B, 0, stream>>>(rowptr, bsum);
  scatter_kernel<<<gE, B, 0, stream>>>(ei, ew, deg, rowptr, cursor, csr_src, csr_w);

  // ---- two DyGr blocks ----
  for (int blk = 0; blk < 2; ++blk) {
    int o = 3 + blk * 9;
    const float* ggc  = p[o + 0];
    const float* gwih = p[o + 1];
    const float* gwhh = p[o + 2];
    const float* gbih = p[o + 3];
    const float* gbhh = p[o + 4];
    const float* lwih = p[o + 5];
    const float* lbih = p[o + 7];
    const float* lbhh = p[o + 8];

    const float* hcur = (blk == 0) ? x : hC;
    float* bufs[2] = {hA, hB};
    int pp = 0;
    for (int l = 0; l < NL; ++l) {
      float* hn = bufs[pp];
      pp ^= 1;
      gemm_m_kernel<<<gT, B, 0, stream>>>(hcur, ggc + (size_t)l * FD * FD, mb);
      gather_kernel<<<gNw, B, 0, stream>>>(rowptr, csr_src, csr_w, mb, agg);
      gru_kernel<<<gT, B, 0, stream>>>(hcur, agg, gwih, gwhh, gbih, gbhh, hn);
      hcur = hn;
    }
    lstm_kernel<<<gT, B, 0, stream>>>(hcur, lwih, lbih, lbhh, (blk == 0) ? hC : mb);
  }

  final_kernel<<<gNw, B, 0, stream>>>(mb, p[21], p[22], (float*)d_out);
}